// GroupedLinear_40596030882515
// MI455X (gfx1250) — compile-verified
//
#include <hip/hip_runtime.h>

typedef __attribute__((ext_vector_type(2))) float v2f;
typedef __attribute__((ext_vector_type(8))) float v8f;

#define B_    4
#define S_    4096
#define IN_F  4096
#define OUT_F 4096
#define NB    128
#define IPB   32
#define OPB   32
#define TOKENS (B_ * S_)

#define WAVES_PER_WG   8
#define TILES_PER_WAVE 8                                   // 16 tokens per tile
#define TOKENS_PER_WG  (WAVES_PER_WG * TILES_PER_WAVE * 16) // 1024
#define NCHUNK         (TOKENS / TOKENS_PER_WG)             // 16

__global__ __launch_bounds__(256) void GroupedLinear_wmma_f32(
    const float* __restrict__ x,     // [TOKENS, IN_F]
    const float* __restrict__ w,     // [NB, IPB, OPB]
    const float* __restrict__ bias,  // [NB, OPB]
    float* __restrict__ out)         // [TOKENS, OUT_F]
{
    const int lane = threadIdx.x & 31;
    const int wave = threadIdx.x >> 5;
    const int n     = blockIdx.x % NB;     // weight block
    const int chunk = blockIdx.x / NB;     // token chunk

    const int half = lane >> 4;   // 0: lanes 0-15, 1: lanes 16-31
    const int l16  = lane & 15;

    // ---- B fragments for block n, register-resident across all tiles ----
    // WMMA f32 16x16x4 B (4x16): VGPR0 = rows K={0 or 2} (by lane half),
    // VGPR1 = rows K={1 or 3}; N = lane%16.
    const float* wn = w + (size_t)n * (IPB * OPB);
    v2f bfr[2][8];
#pragma unroll
    for (int t = 0; t < 2; ++t) {
#pragma unroll
        for (int f = 0; f < 8; ++f) {
            const int krow = 4 * f + 2 * half;
            const int col  = 16 * t + l16;
            bfr[t][f].x = wn[(size_t)krow       * OPB + col];
            bfr[t][f].y = wn[(size_t)(krow + 1) * OPB + col];
        }
    }

    // Bias folded into accumulator init (D layout: col = lane%16 [+16 for tile 1])
    const float bv0 = bias[n * OPB + l16];
    const float bv1 = bias[n * OPB + 16 + l16];

    const int tokBase0 = chunk * TOKENS_PER_WG + wave * (TILES_PER_WAVE * 16);

    // A layout (16x4 f32): lane<16 -> row=l16, K pair {4f,4f+1};
    // lane>=16 -> row=l16, K pair {4f+2,4f+3}. 8-byte aligned b64 loads.
    const float* xp = x + (size_t)(tokBase0 + l16) * IN_F + n * IPB + 2 * half;

    v2f a[8];
#pragma unroll
    for (int f = 0; f < 8; ++f)
        a[f] = *(const v2f*)(xp + 4 * f);

    for (int it = 0; it < TILES_PER_WAVE; ++it) {
        // Prefetch next tile's A fragments (double buffer)
        v2f an[8];
        if (it + 1 < TILES_PER_WAVE) {
            const float* xpn = xp + (size_t)(it + 1) * 16 * IN_F;
#pragma unroll
            for (int f = 0; f < 8; ++f)
                an[f] = *(const v2f*)(xpn + 4 * f);
        }

        v8f acc0 = {bv0, bv0, bv0, bv0, bv0, bv0, bv0, bv0};
        v8f acc1 = {bv1, bv1, bv1, bv1, bv1, bv1, bv1, bv1};

#pragma unroll
        for (int f = 0; f < 8; ++f) {
            acc0 = __builtin_amdgcn_wmma_f32_16x16x4_f32(
                false, a[f], false, bfr[0][f], (short)0, acc0, false, false);
            acc1 = __builtin_amdgcn_wmma_f32_16x16x4_f32(
                false, a[f], false, bfr[1][f], (short)0, acc1, false, false);
        }

        // D layout: row = v + 8*half, col = l16 (+16 for acc1)
        const int tokBase = tokBase0 + it * 16;
        float* op = out + (size_t)(tokBase + 8 * half) * OUT_F + n * OPB + l16;
#pragma unroll
        for (int v = 0; v < 8; ++v) {
            op[(size_t)v * OUT_F]      = acc0[v];
            op[(size_t)v * OUT_F + 16] = acc1[v];
        }

        if (it + 1 < TILES_PER_WAVE) {
#pragma unroll
            for (int f = 0; f < 8; ++f) a[f] = an[f];
        }
    }
}

extern "C" void kernel_launch(void* const* d_in, const int* in_sizes, int n_in,
                              void* d_out, int out_size, void* d_ws, size_t ws_size,
                              hipStream_t stream) {
    const float* x    = (const float*)d_in[0];
    const float* w    = (const float*)d_in[1];
    const float* bias = (const float*)d_in[2];
    float* out        = (float*)d_out;

    dim3 grid(NB * NCHUNK);   // 2048 workgroups
    dim3 block(256);          // 8 waves (wave32)
    GroupedLinear_wmma_f32<<<grid, block, 0, stream>>>(x, w, bias, out);
}